// Sampler_85186381349339
// MI455X (gfx1250) — compile-verified
//
#include <hip/hip_runtime.h>
#include <hip/hip_bf16.h>
#include <stdint.h>

// ---------------------------------------------------------------------------
// Sampler for MI455X (gfx1250): repetition penalty + temperature + top-k mask
// + Gumbel (Threefry-2x32) categorical sample.
// One workgroup per batch row; streaming radix-select; async-to-LDS staging.
// ---------------------------------------------------------------------------

namespace {
constexpr int kV        = 151936;
constexpr int kHistLen  = 200;
constexpr int kNV4      = kV / 4;              // 37984 float4 per row
constexpr int kSeenW    = (kV + 31) / 32;      // 4748 u32 bitmask words
constexpr int kBins     = 4096;                // 12-bit radix histogram
constexpr int kCap      = 4096;                // candidate buffer cap
constexpr int kTpb      = 1024;                // 32 waves (wave32)
constexpr int kIters    = (kNV4 + kTpb - 1) / kTpb;  // 38
}

__device__ __forceinline__ uint32_t rotl32(uint32_t x, int n) {
  return (x << n) | (x >> (32 - n));
}

// JAX threefry2x32, key = [0, 12345] for jax.random.key(12345).
__device__ __forceinline__ void threefry2x32(uint32_t k0, uint32_t k1,
                                             uint32_t x0, uint32_t x1,
                                             uint32_t& o0, uint32_t& o1) {
  const uint32_t ks2 = k0 ^ k1 ^ 0x1BD11BDAu;
  x0 += k0; x1 += k1;
  x0 += x1; x1 = rotl32(x1, 13); x1 ^= x0;
  x0 += x1; x1 = rotl32(x1, 15); x1 ^= x0;
  x0 += x1; x1 = rotl32(x1, 26); x1 ^= x0;
  x0 += x1; x1 = rotl32(x1,  6); x1 ^= x0;
  x0 += k1; x1 += ks2 + 1u;
  x0 += x1; x1 = rotl32(x1, 17); x1 ^= x0;
  x0 += x1; x1 = rotl32(x1, 29); x1 ^= x0;
  x0 += x1; x1 = rotl32(x1, 16); x1 ^= x0;
  x0 += x1; x1 = rotl32(x1, 24); x1 ^= x0;
  x0 += ks2; x1 += k0 + 2u;
  x0 += x1; x1 = rotl32(x1, 13); x1 ^= x0;
  x0 += x1; x1 = rotl32(x1, 15); x1 ^= x0;
  x0 += x1; x1 = rotl32(x1, 26); x1 ^= x0;
  x0 += x1; x1 = rotl32(x1,  6); x1 ^= x0;
  x0 += k0; x1 += k1 + 3u;
  x0 += x1; x1 = rotl32(x1, 17); x1 ^= x0;
  x0 += x1; x1 = rotl32(x1, 29); x1 ^= x0;
  x0 += x1; x1 = rotl32(x1, 16); x1 ^= x0;
  x0 += x1; x1 = rotl32(x1, 24); x1 ^= x0;
  x0 += k1; x1 += ks2 + 4u;
  x0 += x1; x1 = rotl32(x1, 13); x1 ^= x0;
  x0 += x1; x1 = rotl32(x1, 15); x1 ^= x0;
  x0 += x1; x1 = rotl32(x1, 26); x1 ^= x0;
  x0 += x1; x1 = rotl32(x1,  6); x1 ^= x0;
  x0 += ks2; x1 += k0 + 5u;
  o0 = x0; o1 = x1;
}

// Monotonic (total-order) key for finite f32.
__device__ __forceinline__ uint32_t orderKey(float x) {
  uint32_t u = __float_as_uint(x);
  return (u & 0x80000000u) ? ~u : (u | 0x80000000u);
}

__global__ void __launch_bounds__(kTpb, 1)
sampler_kernel(const float* __restrict__ logits,
               const float* __restrict__ temps,
               const float* __restrict__ pens,
               const int*   __restrict__ prev,
               const int*   __restrict__ topk,
               float*       __restrict__ out,
               int nRows) {
  __shared__ uint32_t s_seen[kSeenW];        // 18992 B
  __shared__ uint32_t s_hist[kBins];         // 16384 B
  __shared__ uint32_t s_scan[kTpb];          //  4096 B
  __shared__ float    s_cval[kCap];          // 16384 B
  __shared__ int      s_cidx[kCap];          // 16384 B
  __shared__ float4   s_stage[2][kTpb];      // 32768 B
  __shared__ float    s_redv[kTpb];          //  4096 B
  __shared__ int      s_redi[kTpb];          //  4096 B
  __shared__ int      s_cnt;
  __shared__ int      s_threshBin;
  __shared__ float    s_kth;

  const int b = blockIdx.x;
  const int t = threadIdx.x;
  const int K = topk[0];                     // 50
  const float pen  = pens[b];
  const float temp = temps[b];
  const float*  rowBase  = logits + (size_t)b * kV;
  const float4* rowBase4 = (const float4*)rowBase;

  // ---- Phase 0: clear LDS state -------------------------------------------
  for (int i = t; i < kSeenW; i += kTpb) s_seen[i] = 0u;
  for (int i = t; i < kBins;  i += kTpb) s_hist[i] = 0u;
  if (t == 0) s_cnt = 0;
  __syncthreads();

  // ---- Phase 1: scatter seen-token bitmask --------------------------------
  for (int i = t; i < kHistLen; i += kTpb) {
    int id = prev[b * kHistLen + i];
    if (id >= 0 && id < kV) atomicOr(&s_seen[id >> 5], 1u << (id & 31));
  }
  __syncthreads();

  // ---- Phase 2: HBM pass, async-to-LDS double-buffered, build histogram ---
  const uint32_t ldsA0 = (uint32_t)(uintptr_t)&s_stage[0][t];
  const uint32_t ldsA1 = (uint32_t)(uintptr_t)&s_stage[1][t];
  const uint64_t gbase = (uint64_t)(uintptr_t)rowBase;
  {
    // prologue: stage tile 0
    uint32_t go = (uint32_t)t * 16u;
    asm volatile("global_load_async_to_lds_b128 %0, %1, %2"
                 :: "v"(ldsA0), "v"(go), "s"(gbase) : "memory");
  }
  for (int s = 0; s < kIters; ++s) {
    const int inext = (s + 1) * kTpb + t;
    if (inext < kNV4) {
      // prior DS read of the slot we are about to overwrite must be done
      asm volatile("s_wait_dscnt 0x0" ::: "memory");
      uint32_t go  = (uint32_t)inext * 16u;
      uint32_t lds = ((s + 1) & 1) ? ldsA1 : ldsA0;
      asm volatile("global_load_async_to_lds_b128 %0, %1, %2"
                   :: "v"(lds), "v"(go), "s"(gbase) : "memory");
      asm volatile("s_wait_asynccnt 0x1" ::: "memory");   // current tile ready
    } else {
      asm volatile("s_wait_asynccnt 0x0" ::: "memory");
    }
    const int i = s * kTpb + t;
    if (i < kNV4) {
      float4 vv = s_stage[s & 1][t];
      const float xs[4] = {vv.x, vv.y, vv.z, vv.w};
      const int v0 = i * 4;
      const uint32_t word = s_seen[v0 >> 5];      // 4 consecutive ids share word
#pragma unroll
      for (int j = 0; j < 4; ++j) {
        float x = xs[j];
        if ((word >> ((v0 + j) & 31)) & 1u)
          x = (x > 0.0f) ? (x / pen) : (x * pen);
        x = x / temp;
        atomicAdd(&s_hist[orderKey(x) >> 20], 1u);
      }
    }
  }
  asm volatile("s_wait_asynccnt 0x0" ::: "memory");
  __syncthreads();

  // ---- Phase 3: suffix-scan histogram (descending), locate rank-K bin -----
  uint32_t psum = 0;
#pragma unroll
  for (int j = 0; j < 4; ++j) psum += s_hist[kBins - 1 - (4 * t + j)];
  s_scan[t] = psum;
  __syncthreads();
  for (int off = 1; off < kTpb; off <<= 1) {
    uint32_t add = (t >= off) ? s_scan[t - off] : 0u;
    __syncthreads();
    s_scan[t] += add;
    __syncthreads();
  }
  {
    const uint32_t prevS = (t == 0) ? 0u : s_scan[t - 1];
    if (prevS < (uint32_t)K && s_scan[t] >= (uint32_t)K) {
      uint32_t c = prevS;
#pragma unroll
      for (int j = 0; j < 4; ++j) {
        const int bin = kBins - 1 - (4 * t + j);
        const uint32_t h = s_hist[bin];
        if (c + h >= (uint32_t)K) { s_threshBin = bin; break; }
        c += h;
      }
    }
  }
  __syncthreads();
  const int threshBin = s_threshBin;

  // ---- Phase 4: L2 pass, gather candidates in bins >= threshold -----------
  for (int i = t; i < kNV4; i += kTpb) {
    __builtin_prefetch((const void*)(rowBase4 + i + kTpb), 0, 0);
    float4 vv = rowBase4[i];
    const float xs[4] = {vv.x, vv.y, vv.z, vv.w};
    const int v0 = i * 4;
    const uint32_t word = s_seen[v0 >> 5];
#pragma unroll
    for (int j = 0; j < 4; ++j) {
      float x = xs[j];
      if ((word >> ((v0 + j) & 31)) & 1u)
        x = (x > 0.0f) ? (x / pen) : (x * pen);
      x = x / temp;
      if ((int)(orderKey(x) >> 20) >= threshBin) {
        int pos = atomicAdd(&s_cnt, 1);
        if (pos < kCap) { s_cval[pos] = x; s_cidx[pos] = v0 + j; }
      }
    }
  }
  __syncthreads();
  const int m = min(s_cnt, kCap);

  // ---- Phase 5: exact K-th largest among candidates (counting select) -----
  for (int i = t; i < m; i += kTpb) {
    const float v = s_cval[i];
    int gt = 0, eq = 0;
    for (int j = 0; j < m; ++j) {
      const float w = s_cval[j];
      gt += (w > v);
      eq += (w == v);
    }
    if (gt < K && gt + eq >= K) s_kth = v;   // all writers store same value
  }
  __syncthreads();
  const float kth = s_kth;

  // ---- Phase 6: Gumbel argmax over kept candidates (JAX threefry RNG) -----
  const uint32_t half = (uint32_t)((uint64_t)gridDim.x * (uint64_t)kV / 2u);
  float bestV = -__builtin_inff();
  int   bestI = 0x7fffffff;
  for (int i = t; i < m; i += kTpb) {
    const float v = s_cval[i];
    if (v >= kth) {
      const int idx = s_cidx[i];
      const uint32_t p = (uint32_t)b * (uint32_t)kV + (uint32_t)idx;
      uint32_t c0, c1, o0, o1;
      const bool hi = (p >= half);
      if (hi) { c0 = p - half; c1 = p; } else { c0 = p; c1 = p + half; }
      threefry2x32(0u, 12345u, c0, c1, o0, o1);
      const uint32_t bits = hi ? o1 : o0;
      float u = __uint_as_float((bits >> 9) | 0x3F800000u) - 1.0f;
      u = fmaxf(1.17549435e-38f, u);               // match jax minval=tiny
      const float score = -logf(-logf(u)) + v;
      if (score > bestV || (score == bestV && idx < bestI)) {
        bestV = score; bestI = idx;
      }
    }
  }
  s_redv[t] = bestV; s_redi[t] = bestI;
  __syncthreads();
  for (int off = kTpb / 2; off > 0; off >>= 1) {
    if (t < off) {
      const float ov = s_redv[t + off];
      const int   oi = s_redi[t + off];
      if (ov > s_redv[t] || (ov == s_redv[t] && oi < s_redi[t])) {
        s_redv[t] = ov; s_redi[t] = oi;
      }
    }
    __syncthreads();
  }
  if (t == 0 && b < nRows) out[b] = (float)s_redi[0];
}

extern "C" void kernel_launch(void* const* d_in, const int* in_sizes, int n_in,
                              void* d_out, int out_size, void* d_ws, size_t ws_size,
                              hipStream_t stream) {
  const float* logits = (const float*)d_in[0];
  const float* temps  = (const float*)d_in[1];
  const float* pens   = (const float*)d_in[2];
  const int*   prev   = (const int*)d_in[3];
  const int*   topk   = (const int*)d_in[4];
  float* out = (float*)d_out;
  const int B = in_sizes[1];                 // temperatures has B elements
  sampler_kernel<<<B, kTpb, 0, stream>>>(logits, temps, pens, prev, topk, out, B);
}